// BondEncoding2D_64776696758776
// MI455X (gfx1250) — compile-verified
//
#include <hip/hip_runtime.h>

typedef __attribute__((ext_vector_type(2))) float v2f;
typedef __attribute__((ext_vector_type(4))) float v4f;
typedef __attribute__((ext_vector_type(8))) float v8f;

#define N_NODES 256
#define NN      65536          // 256*256
#define H       32
#define D_MAX   30
#define NEG_SENT (-9999)
#define INF8    255

// ---------------------------------------------------------------------------
// Kernel 1: Floyd-Warshall with predecessors, entirely in LDS.
// dist fits in uint8 (max true distance 30, INF=255; candidate sums compared
// as int so no overflow). pred fits in uint8 (node ids 0..255); the NEG
// sentinel is only ever needed on the diagonal, materialized at write-out.
// One workgroup of 1024 threads owns 64 contiguous cells each (fixed row i
// per thread -> d[i][k] loaded once per k). Row-k / column-k entries provably
// never change during iteration k, so the in-place update is exactly the
// reference's out-of-place semantics; strict '<' preserves tie behavior.
// ---------------------------------------------------------------------------
__global__ __launch_bounds__(1024)
void fw_kernel(const int* __restrict__ aft,      // (2,480) edges
               int* __restrict__ sp_out,         // (256,256) spatial_pos
               int* __restrict__ pred_out)       // (256,256) predecessors
{
    __shared__ unsigned char s_dist[NN];   // 64 KB
    __shared__ unsigned char s_pred[NN];   // 64 KB
    const int t = threadIdx.x;

    for (int c = t; c < NN; c += 1024) {
        int i = c >> 8, j = c & 255;
        s_dist[c] = (i == j) ? 0 : (unsigned char)INF8;
        s_pred[c] = 0;
    }
    __syncthreads();
    if (t < 480) {
        int u = aft[t], v = aft[480 + t];
        s_dist[u * 256 + v] = 1;
        s_dist[v * 256 + u] = 1;
        s_pred[u * 256 + v] = (unsigned char)u;
        s_pred[v * 256 + u] = (unsigned char)v;
    }
    __syncthreads();

    const int cbase = t * 64;
    const int irow  = cbase >> 8;
    const int jbase = cbase & 255;
    for (int k = 0; k < 256; ++k) {
        int dik = (int)s_dist[irow * 256 + k];
        if (dik < INF8) {
            for (int jj = 0; jj < 64; ++jj) {
                int c  = cbase + jj;
                int nd = dik + (int)s_dist[k * 256 + jbase + jj];
                if (nd < (int)s_dist[c]) {
                    s_dist[c] = (unsigned char)nd;
                    s_pred[c] = s_pred[k * 256 + jbase + jj];
                }
            }
        }
        __syncthreads();
    }

    for (int c = t; c < NN; c += 1024) {
        int i = c >> 8, j = c & 255;
        sp_out[c]   = (int)s_dist[c];
        pred_out[c] = (i == j) ? NEG_SENT : (int)s_pred[c];
    }
}

// ---------------------------------------------------------------------------
// Kernel 2: predecessor walk -> bonds_seq[d][r][c] (uint8 bond ids < 101).
// One block per source row r; pred row, atoms, bond_types cached in LDS.
// Faithful to the reference scan (masked walk, 30 fixed steps).
// ---------------------------------------------------------------------------
__global__ __launch_bounds__(256)
void bonds_kernel(const int* __restrict__ pred,        // (256,256)
                  const int* __restrict__ atoms,       // (256,)
                  const int* __restrict__ bond_types,  // (10,10)
                  unsigned char* __restrict__ BS)      // (30,256,256)
{
    __shared__ int s_predr[256];
    __shared__ int s_atoms[256];
    __shared__ int s_bt[100];
    const int r = blockIdx.x;
    const int c = threadIdx.x;
    s_predr[c] = pred[r * 256 + c];
    s_atoms[c] = atoms[c];
    if (c < 100) s_bt[c] = bond_types[c];
    __syncthreads();

    int  to   = c;
    bool mask = (s_predr[c] != NEG_SENT);
    for (int d = 0; d < D_MAX; ++d) {
        int inner = mask ? s_predr[to] : 0;
        int b     = mask ? s_bt[s_atoms[to] * 10 + s_atoms[inner]] : 0;
        BS[d * NN + r * 256 + c] = (unsigned char)b;
        mask = mask && (inner != r);
        to   = inner;
    }
}

// ---------------------------------------------------------------------------
// Kernel 3: the scrambled-reshape batched GEMM + reductions, in f32 WMMA.
//   s[p,k] = sum_{d2<30} sum_{h<32} edge_weight[BS[(d2*256+a)%30][(d2*256+a)/30][b], h]
//                                   * w[d2,h,k]        (p = a*256+b)
//   phi_edge[k,a,b] = s / max(dist,1);  phi_spd[k,p] = spd_weight[dist[p],k]
// One wave per 16-row strip of p, N=32 via two accumulators; 480 chained
// v_wmma_f32_16x16x4_f32 per wave (16 static, d2-loop not unrolled).
// Fragment layouts per ISA 7.12.2 (f32 16x16x4):
//   A: lane M=lane&15, VGPR0=K(base), VGPR1=K(base+1), base=lane[4]*2
//   B: lane N=lane&15, VGPR0=B[base,N], VGPR1=B[base+1,N]
//   D: VGPR r -> M=r+lane[4]*8, N=lane&15
// w is staged into LDS PRE-PERMUTED so a lane's 4 B values per chunk,
//   (K,col),(K+1,col),(K,col+16),(K+1,col+16),
// are 16B-contiguous -> one aligned ds_load_b128 yields both B fragments
// on even VGPR pairs (no repacking movs).
//   dest(d2,h,k) = d2*1024 + ((h>>1)*16 + (k&15))*4 + (k>>4)*2 + (h&1)
// ---------------------------------------------------------------------------
__global__ __launch_bounds__(256)
void edge_gemm_kernel(const unsigned char* __restrict__ BS,         // (30,256,256)
                      const float* __restrict__ edge_weight,        // (101,32)
                      const float* __restrict__ edge_dis_weight,    // first 30*32*32 used
                      const float* __restrict__ spd_weight,         // (512,32)
                      const int*   __restrict__ sp,                 // (256,256)
                      float* __restrict__ out)                      // phi_spd | phi_edge
{
    __shared__ float s_w[D_MAX * H * H];  // 30720 floats = 120 KB (permuted)
    __shared__ float s_ew[101 * H];       //  3232 floats = 12.6 KB
    const int tid = threadIdx.x;
    for (int i = tid; i < D_MAX * H * H; i += 256) {
        const int d2  = i >> 10;          // i / 1024
        const int rem = i & 1023;
        const int h   = rem >> 5;         // row in w[d2]
        const int k   = rem & 31;         // col in w[d2]
        const int dst = (d2 << 10) + (((h >> 1) * 16 + (k & 15)) << 2)
                        + ((k >> 4) << 1) + (h & 1);
        s_w[dst] = edge_dis_weight[i];
    }
    for (int i = tid; i < 101 * H; i += 256) s_ew[i] = edge_weight[i];
    __syncthreads();

    const int lane   = tid & 31;
    const int wave   = tid >> 5;
    const int wid    = blockIdx.x * 8 + wave;   // 0..4095, exact
    const int p_base = wid * 16;
    const int a      = p_base >> 8;
    const int b0     = p_base & 255;

    const int m   = lane & 15;          // M for A, N for B
    const int hi  = lane >> 4;          // 0 or 1
    const int khi = hi << 1;            // K sub-offset 0 or 2

    v8f acc0 = {};
    v8f acc1 = {};

    for (int d2 = 0; d2 < D_MAX; ++d2) {
        const int t2     = d2 * 256 + a;
        const int d      = t2 % D_MAX;
        const int j      = t2 / D_MAX;
        const int bidx   = (int)BS[d * NN + j * 256 + b0 + m];
        const int rowoff = bidx * H;
        // per-lane base into the permuted w block for this d2:
        //   chunk h0 -> group (h0>>1)*16 + hi*8 + m, 4 floats each
        const int wbase = (d2 << 10) + (m << 2) + (hi << 5);
#pragma unroll
        for (int h0 = 0; h0 < H; h0 += 4) {
            v2f af;
            af.x = s_ew[rowoff + h0 + khi];
            af.y = s_ew[rowoff + h0 + khi + 1];
            // ((h0+khi)>>1)*64 == (h0>>1)*64 + hi*64; wbase adds only hi<<5 = hi*32
            const v4f q = *(const v4f*)&s_w[wbase + (h0 << 5)];
            v2f bf0, bf1;
            bf0.x = q.x; bf0.y = q.y;
            bf1.x = q.z; bf1.y = q.w;
            acc0 = __builtin_amdgcn_wmma_f32_16x16x4_f32(
                       false, af, false, bf0, (short)0, acc0, false, false);
            acc1 = __builtin_amdgcn_wmma_f32_16x16x4_f32(
                       false, af, false, bf1, (short)0, acc1, false, false);
        }
    }

    // Epilogue: divide by max(dist,1), write phi_edge; fuse phi_spd gather.
    const int Mbase = hi * 8;
#pragma unroll
    for (int r = 0; r < 8; ++r) {
        const int p  = p_base + Mbase + r;
        const int dp = sp[p];
        const float spf = (dp == 0) ? 1.0f : (float)dp;
        out[2097152 + m * NN + p]        = acc0[r] / spf;   // phi_edge[k=m]
        out[2097152 + (m + 16) * NN + p] = acc1[r] / spf;   // phi_edge[k=m+16]
        out[m * NN + p]        = spd_weight[dp * H + m];        // phi_spd[k=m]
        out[(m + 16) * NN + p] = spd_weight[dp * H + m + 16];   // phi_spd[k=m+16]
    }
}

// ---------------------------------------------------------------------------
extern "C" void kernel_launch(void* const* d_in, const int* in_sizes, int n_in,
                              void* d_out, int out_size, void* d_ws, size_t ws_size,
                              hipStream_t stream)
{
    (void)in_sizes; (void)n_in; (void)out_size; (void)ws_size;
    const int*   atoms  = (const int*)d_in[0];
    const int*   aft    = (const int*)d_in[1];   // (2,480)
    const int*   bt     = (const int*)d_in[2];   // (10,10)
    const float* spd_w  = (const float*)d_in[3]; // (512,32)
    const float* edge_w = (const float*)d_in[4]; // (101,32)
    const float* edw    = (const float*)d_in[5]; // (524288,1)
    float* out = (float*)d_out;

    // workspace layout: sp (256KB int) | pred (256KB int) | BS (1.92MB u8)
    int* sp   = (int*)d_ws;
    int* pred = sp + NN;
    unsigned char* BS = (unsigned char*)(pred + NN);

    fw_kernel<<<1, 1024, 0, stream>>>(aft, sp, pred);
    bonds_kernel<<<256, 256, 0, stream>>>(pred, atoms, bt, BS);
    edge_gemm_kernel<<<512, 256, 0, stream>>>(BS, edge_w, edw, spd_w, sp, out);
}